// ModalityFusionTopic_61804579389733
// MI455X (gfx1250) — compile-verified
//
#include <hip/hip_runtime.h>
#include <hip/hip_bf16.h>
#include <math.h>

// ---------------------------------------------------------------------------
// Problem constants (from reference)
// ---------------------------------------------------------------------------
#define BS        32
#define NUM_R     10
#define NUM_P     36
#define SL_Q      20
#define IMG_FEAT  2048
#define TOPIC_DIM 600
#define LSTM      512
#define CAT       2648        // IMG_FEAT + TOPIC_DIM
#define BB        (BS*NUM_R)  // 320 fused batch

typedef __attribute__((ext_vector_type(2))) float v2f;
typedef __attribute__((ext_vector_type(8))) float v8f;

// f32 WMMA: D(16x16) = A(16x4) * B(4x16) + C
__device__ __forceinline__ v8f wmma_f32(v2f a, v2f b, v8f c) {
    return __builtin_amdgcn_wmma_f32_16x16x4_f32(
        false, a, false, b, (short)0, c, false, false);
}

__device__ __forceinline__ v2f ld2(const float* __restrict__ base, unsigned off) {
    return *(const v2f*)(base + off);
}

// ---------------------------------------------------------------------------
// Kernel 1+2: GatedTrans  Y = tanh(X@Wy + by) * leaky_relu(X@Wg + bg)
// 32x32 tile per wave for BOTH GEMMs (A fragment shared).
// K,N compile-time so B strides fold into load immediate offsets.
// K % 8 == 0, M % 32 == 0, N % 32 == 0 at both call sites.
// ---------------------------------------------------------------------------
template<int K, int N>
__global__ __launch_bounds__(32)
void gated_gemm_k(const float* __restrict__ X,
                  const float* __restrict__ Wy, const float* __restrict__ by,
                  const float* __restrict__ Wg, const float* __restrict__ bg,
                  float* __restrict__ Y) {
    const int lane = threadIdx.x;
    const int l = lane & 15;
    const int h = lane >> 4;
    const int m0 = blockIdx.x << 5;
    const int n0 = blockIdx.y << 5;

    unsigned aOff0 = (unsigned)(m0 + l) * K + (h << 1);
    unsigned aOff1 = aOff0 + 16u * K;
    unsigned bOff  = (unsigned)(h << 1) * N + n0 + l;   // +16 for second N tile

    v8f cy0 = {}, cy1 = {}, cy2 = {}, cy3 = {};
    v8f cg0 = {}, cg1 = {}, cg2 = {}, cg3 = {};

    for (int k0 = 0; k0 < K; k0 += 8) {
#pragma unroll
        for (int u = 0; u < 2; ++u) {
            const unsigned ao = 4u * u;
            const unsigned bo = bOff + 4u * u * N;
            v2f a0 = ld2(X, aOff0 + ao);
            v2f a1 = ld2(X, aOff1 + ao);
            v2f b0 = { Wy[bo],      Wy[bo + N]      };
            v2f b1 = { Wy[bo + 16], Wy[bo + N + 16] };
            v2f g0 = { Wg[bo],      Wg[bo + N]      };
            v2f g1 = { Wg[bo + 16], Wg[bo + N + 16] };
            cy0 = wmma_f32(a0, b0, cy0);  cy1 = wmma_f32(a0, b1, cy1);
            cy2 = wmma_f32(a1, b0, cy2);  cy3 = wmma_f32(a1, b1, cy3);
            cg0 = wmma_f32(a0, g0, cg0);  cg1 = wmma_f32(a0, g1, cg1);
            cg2 = wmma_f32(a1, g0, cg2);  cg3 = wmma_f32(a1, g1, cg3);
        }
        aOff0 += 8; aOff1 += 8; bOff += 8u * N;
    }

    const float biy0 = by[n0 + l],      big0 = bg[n0 + l];
    const float biy1 = by[n0 + 16 + l], big1 = bg[n0 + 16 + l];
#pragma unroll
    for (int v = 0; v < 8; ++v) {
        const unsigned r0 = (unsigned)(m0 + v + (h << 3)) * N + n0 + l;
        const unsigned r1 = r0 + 16u * N;
        float yv, gv;
        yv = cy0[v] + biy0; gv = cg0[v] + big0; gv = gv > 0.f ? gv : 0.01f * gv;
        Y[r0]      = tanhf(yv) * gv;
        yv = cy1[v] + biy1; gv = cg1[v] + big1; gv = gv > 0.f ? gv : 0.01f * gv;
        Y[r0 + 16] = tanhf(yv) * gv;
        yv = cy2[v] + biy0; gv = cg2[v] + big0; gv = gv > 0.f ? gv : 0.01f * gv;
        Y[r1]      = tanhf(yv) * gv;
        yv = cy3[v] + biy1; gv = cg3[v] + big1; gv = gv > 0.f ? gv : 0.01f * gv;
        Y[r1 + 16] = tanhf(yv) * gv;
    }
}

// ---------------------------------------------------------------------------
// Kernel 3: attention.  Per fused batch b (320 blocks):
//   scores(36,20) = v_feat(bs) @ topic_feat(b)^T ; mask ; softmax
//   vt_att(36,600) = scores @ topic_flat(b)
// ---------------------------------------------------------------------------
__global__ __launch_bounds__(128)
void attn_k(const float* __restrict__ vfeat,       // (BS*NUM_P, LSTM)
            const float* __restrict__ tfeat,       // (BB*SL_Q, LSTM)
            const float* __restrict__ topic,       // (BB*SL_Q, TOPIC_DIM)
            const int*   __restrict__ qmask,       // (BB, SL_Q)
            float* __restrict__ vt_att) {          // (BB*NUM_P, TOPIC_DIM)
    const int b  = blockIdx.x;
    const int bs = b / NUM_R;
    const int tid = threadIdx.x;

    __shared__ float sc[NUM_P * SL_Q];

    for (int i = tid; i < NUM_P * SL_Q; i += 128) {
        const int p = i / SL_Q, s = i % SL_Q;
        const float* vp = vfeat + ((size_t)bs * NUM_P + p) * LSTM;
        const float* tp = tfeat + ((size_t)b * SL_Q + s) * LSTM;
        float acc = 0.0f;
        for (int x = 0; x < LSTM; ++x) acc += vp[x] * tp[x];
        const float mv = (float)qmask[b * SL_Q + s];
        sc[i] = acc * mv + (mv - 1.0f) * 10000.0f;
    }
    __syncthreads();

    if (tid < NUM_P) {
        float mx = -INFINITY;
        for (int s = 0; s < SL_Q; ++s) mx = fmaxf(mx, sc[tid * SL_Q + s]);
        float sum = 0.0f;
        for (int s = 0; s < SL_Q; ++s) {
            const float e = __expf(sc[tid * SL_Q + s] - mx);
            sc[tid * SL_Q + s] = e;
            sum += e;
        }
        const float inv = 1.0f / sum;
        for (int s = 0; s < SL_Q; ++s) sc[tid * SL_Q + s] *= inv;
    }
    __syncthreads();

    for (int i = tid; i < NUM_P * TOPIC_DIM; i += 128) {
        const int p = i / TOPIC_DIM, t = i % TOPIC_DIM;
        const float* tf = topic + (size_t)b * SL_Q * TOPIC_DIM + t;
        float acc = 0.0f;
#pragma unroll
        for (int s = 0; s < SL_Q; ++s) acc += sc[p * SL_Q + s] * tf[(size_t)s * TOPIC_DIM];
        vt_att[((size_t)b * NUM_P + p) * TOPIC_DIM + t] = acc;
    }
}

// ---------------------------------------------------------------------------
// Kernel 4: h1 = relu(concat(img_bcast, vt_att) @ W1 + b1)
// 64x32 tile per wave; two k-segments (img K=2048 step 16, att K=600 step 8).
// Virtual concat: img never materialized (saves 94 MB of HBM reads).
// M=11520 (%64), N=2648 (NOT %32: clamp loads, mask stores).
// ---------------------------------------------------------------------------
__global__ __launch_bounds__(32)
void mlp1_k(const float* __restrict__ img,      // (BS, NUM_P, IMG_FEAT)
            const float* __restrict__ att,      // (BB*NUM_P, TOPIC_DIM)
            const float* __restrict__ W1, const float* __restrict__ b1,
            float* __restrict__ h1) {           // (BB*NUM_P, CAT)
    const int lane = threadIdx.x;
    const int l = lane & 15;
    const int h = lane >> 4;
    const int m0 = blockIdx.x << 6;             // 64 rows
    const int n0 = blockIdx.y << 5;             // 32 cols

    unsigned iOff[4], tOff[4];
#pragma unroll
    for (int i = 0; i < 4; ++i) {
        const int m  = m0 + (i << 4) + l;
        const int b  = m / NUM_P;
        const int p  = m % NUM_P;
        const int bs = b / NUM_R;
        iOff[i] = (unsigned)(bs * NUM_P + p) * IMG_FEAT  + (h << 1);
        tOff[i] = (unsigned)(b  * NUM_P + p) * TOPIC_DIM + (h << 1);
    }

    const int  na = n0 + l,  nb = n0 + 16 + l;
    const bool oka = (na < CAT), okb = (nb < CAT);
    unsigned wA = (unsigned)(h << 1) * CAT + (oka ? na : (CAT - 1));
    unsigned wB = (unsigned)(h << 1) * CAT + (okb ? nb : (CAT - 1));

    v8f c00 = {}, c01 = {}, c10 = {}, c11 = {},
        c20 = {}, c21 = {}, c30 = {}, c31 = {};

    // ---- segment 1: img columns, K = 2048, step 16, unroll 4 ----
    for (int it = 0; it < IMG_FEAT / 16; ++it) {
#pragma unroll
        for (int u = 0; u < 4; ++u) {
            const unsigned ao = 4u * u;
            const unsigned wo = 4u * u * CAT;
            v2f a0 = ld2(img, iOff[0] + ao);
            v2f a1 = ld2(img, iOff[1] + ao);
            v2f a2 = ld2(img, iOff[2] + ao);
            v2f a3 = ld2(img, iOff[3] + ao);
            v2f b0 = { W1[wA + wo], W1[wA + wo + CAT] };
            v2f b1 = { W1[wB + wo], W1[wB + wo + CAT] };
            c00 = wmma_f32(a0, b0, c00);  c01 = wmma_f32(a0, b1, c01);
            c10 = wmma_f32(a1, b0, c10);  c11 = wmma_f32(a1, b1, c11);
            c20 = wmma_f32(a2, b0, c20);  c21 = wmma_f32(a2, b1, c21);
            c30 = wmma_f32(a3, b0, c30);  c31 = wmma_f32(a3, b1, c31);
        }
        iOff[0] += 16; iOff[1] += 16; iOff[2] += 16; iOff[3] += 16;
        wA += 16u * CAT; wB += 16u * CAT;
    }

    // ---- segment 2: vt_att columns, K = 600, step 8, unroll 2 ----
    for (int it = 0; it < TOPIC_DIM / 8; ++it) {
#pragma unroll
        for (int u = 0; u < 2; ++u) {
            const unsigned ao = 4u * u;
            const unsigned wo = 4u * u * CAT;
            v2f a0 = ld2(att, tOff[0] + ao);
            v2f a1 = ld2(att, tOff[1] + ao);
            v2f a2 = ld2(att, tOff[2] + ao);
            v2f a3 = ld2(att, tOff[3] + ao);
            v2f b0 = { W1[wA + wo], W1[wA + wo + CAT] };
            v2f b1 = { W1[wB + wo], W1[wB + wo + CAT] };
            c00 = wmma_f32(a0, b0, c00);  c01 = wmma_f32(a0, b1, c01);
            c10 = wmma_f32(a1, b0, c10);  c11 = wmma_f32(a1, b1, c11);
            c20 = wmma_f32(a2, b0, c20);  c21 = wmma_f32(a2, b1, c21);
            c30 = wmma_f32(a3, b0, c30);  c31 = wmma_f32(a3, b1, c31);
        }
        tOff[0] += 8; tOff[1] += 8; tOff[2] += 8; tOff[3] += 8;
        wA += 8u * CAT; wB += 8u * CAT;
    }

    const float bia = oka ? b1[na] : 0.0f;
    const float bib = okb ? b1[nb] : 0.0f;
    const v8f* cc0[4] = { &c00, &c10, &c20, &c30 };
    const v8f* cc1[4] = { &c01, &c11, &c21, &c31 };
#pragma unroll
    for (int i = 0; i < 4; ++i) {
#pragma unroll
        for (int v = 0; v < 8; ++v) {
            const unsigned row = m0 + (i << 4) + v + (h << 3);
            if (oka) h1[row * (unsigned)CAT + na] = fmaxf((*cc0[i])[v] + bia, 0.0f);
            if (okb) h1[row * (unsigned)CAT + nb] = fmaxf((*cc1[i])[v] + bib, 0.0f);
        }
    }
}

// ---------------------------------------------------------------------------
// Kernel 5: out = relu(h1 @ W2 + b2).  64x32 tile per wave.
// M=11520 (%64), K=2648 (%8), N=2048 (%32).
// ---------------------------------------------------------------------------
__global__ __launch_bounds__(32)
void mlp2_k(const float* __restrict__ h1,
            const float* __restrict__ W2, const float* __restrict__ b2,
            float* __restrict__ out) {          // (BB*NUM_P, IMG_FEAT)
    const int lane = threadIdx.x;
    const int l = lane & 15;
    const int h = lane >> 4;
    const int m0 = blockIdx.x << 6;
    const int n0 = blockIdx.y << 5;

    unsigned aOff0 = (unsigned)(m0      + l) * CAT + (h << 1);
    unsigned aOff1 = (unsigned)(m0 + 16 + l) * CAT + (h << 1);
    unsigned aOff2 = (unsigned)(m0 + 32 + l) * CAT + (h << 1);
    unsigned aOff3 = (unsigned)(m0 + 48 + l) * CAT + (h << 1);
    unsigned wOff  = (unsigned)(h << 1) * IMG_FEAT + n0 + l;

    v8f c00 = {}, c01 = {}, c10 = {}, c11 = {},
        c20 = {}, c21 = {}, c30 = {}, c31 = {};

    for (int k0 = 0; k0 < CAT; k0 += 8) {
#pragma unroll
        for (int u = 0; u < 2; ++u) {
            const unsigned ao = 4u * u;
            const unsigned wo = wOff + 4u * u * IMG_FEAT;
            v2f a0 = ld2(h1, aOff0 + ao);
            v2f a1 = ld2(h1, aOff1 + ao);
            v2f a2 = ld2(h1, aOff2 + ao);
            v2f a3 = ld2(h1, aOff3 + ao);
            v2f b0 = { W2[wo],      W2[wo + IMG_FEAT]      };
            v2f b1 = { W2[wo + 16], W2[wo + IMG_FEAT + 16] };
            c00 = wmma_f32(a0, b0, c00);  c01 = wmma_f32(a0, b1, c01);
            c10 = wmma_f32(a1, b0, c10);  c11 = wmma_f32(a1, b1, c11);
            c20 = wmma_f32(a2, b0, c20);  c21 = wmma_f32(a2, b1, c21);
            c30 = wmma_f32(a3, b0, c30);  c31 = wmma_f32(a3, b1, c31);
        }
        aOff0 += 8; aOff1 += 8; aOff2 += 8; aOff3 += 8;
        wOff += 8u * IMG_FEAT;
    }

    const float bia = b2[n0 + l];
    const float bib = b2[n0 + 16 + l];
    const v8f* cc0[4] = { &c00, &c10, &c20, &c30 };
    const v8f* cc1[4] = { &c01, &c11, &c21, &c31 };
#pragma unroll
    for (int i = 0; i < 4; ++i) {
#pragma unroll
        for (int v = 0; v < 8; ++v) {
            const unsigned row = m0 + (i << 4) + v + (h << 3);
            out[row * (unsigned)IMG_FEAT + n0 + l]      = fmaxf((*cc0[i])[v] + bia, 0.0f);
            out[row * (unsigned)IMG_FEAT + n0 + 16 + l] = fmaxf((*cc1[i])[v] + bib, 0.0f);
        }
    }
}

// ---------------------------------------------------------------------------
// Host launcher
// ---------------------------------------------------------------------------
extern "C" void kernel_launch(void* const* d_in, const int* in_sizes, int n_in,
                              void* d_out, int out_size, void* d_ws, size_t ws_size,
                              hipStream_t stream) {
    const float* img    = (const float*)d_in[0];   // (32,36,2048)
    const float* topic  = (const float*)d_in[1];   // (32,10,20,600)
    const int*   qmask  = (const int*)  d_in[2];   // (32,10,20)
    const float* Wy_img = (const float*)d_in[3];
    const float* by_img = (const float*)d_in[4];
    const float* Wg_img = (const float*)d_in[5];
    const float* bg_img = (const float*)d_in[6];
    const float* Wy_t   = (const float*)d_in[7];
    const float* by_t   = (const float*)d_in[8];
    const float* Wg_t   = (const float*)d_in[9];
    const float* bg_t   = (const float*)d_in[10];
    const float* W1     = (const float*)d_in[11];
    const float* b1     = (const float*)d_in[12];
    const float* W2     = (const float*)d_in[13];
    const float* b2     = (const float*)d_in[14];
    float* out = (float*)d_out;

    // workspace layout (floats); total ~158 MB
    float* ws      = (float*)d_ws;
    float* v_feat  = ws;                                       // 1152*512
    float* t_feat  = v_feat + (size_t)BS * NUM_P * LSTM;       // 6400*512
    float* vt_att  = t_feat + (size_t)BB * SL_Q * LSTM;        // 11520*600
    float* h1      = vt_att + (size_t)BB * NUM_P * TOPIC_DIM;  // 11520*2648

    // 1) image GatedTrans: M=1152, K=2048, N=512
    gated_gemm_k<IMG_FEAT, LSTM>
        <<<dim3((BS * NUM_P) / 32, LSTM / 32), 32, 0, stream>>>(
        img, Wy_img, by_img, Wg_img, bg_img, v_feat);

    // 2) topic GatedTrans: M=6400, K=600, N=512
    gated_gemm_k<TOPIC_DIM, LSTM>
        <<<dim3((BB * SL_Q) / 32, LSTM / 32), 32, 0, stream>>>(
        topic, Wy_t, by_t, Wg_t, bg_t, t_feat);

    // 3) attention + softmax + weighted topic aggregation
    attn_k<<<dim3(BB), 128, 0, stream>>>(v_feat, t_feat, topic, qmask, vt_att);

    // 4) MLP layer 1 (virtual concat A): M=11520, K=2648, N=2648
    mlp1_k<<<dim3((BB * NUM_P) / 64, (CAT + 31) / 32), 32, 0, stream>>>(
        img, vt_att, W1, b1, h1);

    // 5) MLP layer 2: M=11520, K=2648, N=2048
    mlp2_k<<<dim3((BB * NUM_P) / 64, IMG_FEAT / 32), 32, 0, stream>>>(
        h1, W2, b2, out);
}